// mLSTMBackend_40063454937567
// MI455X (gfx1250) — compile-verified
//
#include <hip/hip_runtime.h>

// mLSTM forward, chunkwise-parallel reformulation, fp32 WMMA + TDM async tensor
// loads (CDNA5 / gfx1250). B=4, NH=8, S=2048, DH=64.
// One workgroup per (b,h) head; 16 waves; chunk L=64; double-buffered LDS.

typedef float v2f __attribute__((ext_vector_type(2)));
typedef float v8f __attribute__((ext_vector_type(8)));

#define LSTR 68           // LDS row stride (floats): 16B-aligned rows, conflict-free columns
#define EPSV 1e-6f

#if __has_builtin(__builtin_amdgcn_tensor_load_to_lds) && __has_builtin(__builtin_amdgcn_s_wait_tensorcnt)
#define HAVE_TDM 1
#else
#define HAVE_TDM 0
#endif

__device__ __forceinline__ v8f wmma_f32(v2f a, v2f b, v8f c) {
  // D = A(16x4,f32) * B(4x16,f32) + C(16x16,f32)
  return __builtin_amdgcn_wmma_f32_16x16x4_f32(
      /*neg_a=*/false, a, /*neg_b=*/false, b,
      /*c_mod=*/(short)0, c, /*reuse_a=*/false, /*reuse_b=*/false);
}

#if HAVE_TDM
typedef unsigned int u32x4 __attribute__((ext_vector_type(4)));
typedef int          i32x4 __attribute__((ext_vector_type(4)));
typedef int          i32x8 __attribute__((ext_vector_type(8)));
typedef __attribute__((address_space(3))) float lds_f;

// TDM: DMA a 64x64 f32 tile (global row stride 64 floats) into LDS with a
// hardware-padded row stride of 68 floats (pad 4 DWORDs after every 64 DWORDs).
// D# layout per CDNA5 ISA ch.8.3-8.5.
__device__ __forceinline__ void tdm_load_64x64(const float* gsrc, float* ldst) {
  unsigned long long ga = (unsigned long long)(const void*)gsrc;
  unsigned lofs = (unsigned)(unsigned long long)(lds_f*)ldst;   // LDS byte offset
  u32x4 g0;
  g0.x = 1u;                                                    // count=1 (valid), user mode
  g0.y = lofs;                                                  // lds_addr
  g0.z = (unsigned)ga;                                          // global_addr[31:0]
  g0.w = (unsigned)((ga >> 32) & 0x01FFFFFFu) | (2u << 30);     // global_addr[56:32], type=2
  i32x8 g1;
  g1[0] = (2 << 16)    // data_size = 4B
        | (1 << 20)    // pad_enable
        | (5 << 22)    // pad_interval = 64 DWORDs
        | (3 << 25);   // pad_amount   = 4 DWORDs  -> LDS row stride 68 floats
  g1[1] = (64 << 16);  // tensor_dim0[15:0] = 64 (bits 79:48, low half here)
  g1[2] = (64 << 16);  // tensor_dim1[15:0] = 64 (bits 111:80, low half here)
  g1[3] = (64 << 16);  // tensor_dim1[31:16]=0 | tile_dim0 = 64
  g1[4] = 64;          // tile_dim1 = 64 | tile_dim2 = 0
  g1[5] = 64;          // tensor_dim0_stride[31:0] = 64 elements
  g1[6] = 0;
  g1[7] = 0;
  i32x4 z4 = {0, 0, 0, 0};
#if __clang_major__ >= 23
  i32x8 z8 = {0, 0, 0, 0, 0, 0, 0, 0};
  __builtin_amdgcn_tensor_load_to_lds(g0, g1, z4, z4, z8, 0);
#else
  __builtin_amdgcn_tensor_load_to_lds(g0, g1, z4, z4, 0);
#endif
}
#endif

__global__ __launch_bounds__(512, 1) void mlstm_fwd(
    const float* __restrict__ q, const float* __restrict__ k,
    const float* __restrict__ v, const float* __restrict__ ig,
    const float* __restrict__ fg, float* __restrict__ out, int S)
{
  __shared__ float Qs[2][64 * LSTR];
  __shared__ float Ks[2][64 * LSTR];
  __shared__ float Vs[2][64 * LSTR];
  __shared__ float Ps[64 * LSTR];   // D ∘ (Q K^T)
  __shared__ float Cs[64 * LSTR];   // scaled covariance state C*exp(-m)
  __shared__ float iL[64], fL[64], bL[64], aL[64], mL[64], gI[64];
  __shared__ float dnL[64], uL[64], nS[64];
  __shared__ float mPrev[1];

  const int tid  = threadIdx.x;
  const int lane = tid & 31;
  const int wv   = tid >> 5;        // wave 0..15
  const int tm   = wv >> 2;         // tile row (0..3)
  const int tn   = wv & 3;          // tile col (0..3)
  const int kh   = lane >> 4;       // lane half (0/1)
  const int l16  = lane & 15;

  const size_t headOff = (size_t)blockIdx.x * (size_t)S * 64;
  const float* qB = q  + headOff;
  const float* kB = k  + headOff;
  const float* vB = v  + headOff;
  const float* iB = ig + (size_t)blockIdx.x * S;
  const float* fB = fg + (size_t)blockIdx.x * S;
  float*       oB = out + headOff;

#if HAVE_TDM
  // Kick chunk 0's DMA immediately; it overlaps the state init below.
  if (wv == 0) {
    tdm_load_64x64(qB, Qs[0]);
    tdm_load_64x64(kB, Ks[0]);
    tdm_load_64x64(vB, Vs[0]);
  }
#endif

  // init state: C=0, n=1, m=0  (reference initial carry, already in scaled form)
  for (int idx = tid; idx < 64 * LSTR; idx += 512) Cs[idx] = 0.f;
  if (tid < 64) nS[tid] = 1.0f;
  if (tid == 0) mPrev[0] = 0.f;

  int p = 0;
  for (int c0 = 0; c0 < S; c0 += 64, p ^= 1) {
    const float* Qb = Qs[p];
    const float* Kb = Ks[p];
    const float* Vb = Vs[p];

#if HAVE_TDM
    if (wv == 0) __builtin_amdgcn_s_wait_tensorcnt(0);  // chunk c0's tiles in LDS
#else
    // ---- synchronous staging fallback ----
    for (int r = tid; r < 64 * 16; r += 512) {
      int row = r >> 4, c4 = (r & 15) << 2;
      float4 xq = *(const float4*)(qB + (size_t)(c0 + row) * 64 + c4);
      float4 xk = *(const float4*)(kB + (size_t)(c0 + row) * 64 + c4);
      float4 xv = *(const float4*)(vB + (size_t)(c0 + row) * 64 + c4);
      *(float4*)&Qs[p][row * LSTR + c4] = xq;
      *(float4*)&Ks[p][row * LSTR + c4] = xk;
      *(float4*)&Vs[p][row * LSTR + c4] = xv;
    }
    if (c0 + 64 < S && tid < 64) {
      __builtin_prefetch(qB + (size_t)(c0 + 64 + tid) * 64, 0, 3);
      __builtin_prefetch(kB + (size_t)(c0 + 64 + tid) * 64, 0, 3);
      __builtin_prefetch(vB + (size_t)(c0 + 64 + tid) * 64, 0, 3);
    }
#endif
    if (tid < 64) {
      iL[tid] = iB[c0 + tid];
      fL[tid] = fB[c0 + tid];
      if (c0 + 64 < S) {
        __builtin_prefetch(iB + c0 + 64 + tid, 0, 3);
        __builtin_prefetch(fB + c0 + 64 + tid, 0, 3);
      }
    }
    __syncthreads();

#if HAVE_TDM
    // Issue next chunk's DMA now: overlaps with all compute phases below.
    if (wv == 0 && c0 + 64 < S) {
      tdm_load_64x64(qB + (size_t)(c0 + 64) * 64, Qs[p ^ 1]);
      tdm_load_64x64(kB + (size_t)(c0 + 64) * 64, Ks[p ^ 1]);
      tdm_load_64x64(vB + (size_t)(c0 + 64) * 64, Vs[p ^ 1]);
    }
#endif

    // ---- P1: serial gate scan (length 64): b=cumsum(f), m_t=max(m+f,i) ----
    if (tid == 0) {
      float mp = mPrev[0];
      float b = 0.f, m = mp;
      for (int t = 0; t < 64; ++t) {
        float ft = fL[t], it = iL[t];
        b += ft;
        m = fmaxf(m + ft, it);
        bL[t] = b; mL[t] = m;
        aL[t] = it - b;
        gI[t] = __expf(mp + b - m);   // inter-chunk decay, exponent <= 0
      }
      mPrev[0] = mL[63];
    }
    __syncthreads();

    // ---- P2: S = Q K^T on lower-triangular tiles, apply decay -> P ----
    if (tn <= tm) {
      v8f acc = {0.f, 0.f, 0.f, 0.f, 0.f, 0.f, 0.f, 0.f};
      #pragma unroll
      for (int kd = 0; kd < 16; ++kd) {
        int kk = kd * 4 + 2 * kh;
        v2f a = *(const v2f*)&Qb[(tm * 16 + l16) * LSTR + kk];   // A = Q
        v2f b = *(const v2f*)&Kb[(tn * 16 + l16) * LSTR + kk];   // B = K^T
        acc = wmma_f32(a, b, acc);
      }
      #pragma unroll
      for (int r = 0; r < 8; ++r) {
        int t = tm * 16 + r + 8 * kh;
        int s = tn * 16 + l16;
        float w = (s <= t) ? __expf(aL[s] + bL[t] - mL[t]) : 0.f;
        Ps[t * LSTR + s] = acc[r] * w;
      }
    }
    __syncthreads();

    // ---- P3: rowsum(P), reciprocal denominator, and u weights ----
    if (tid < 64) {
      int t = tid;
      float rs = 0.f;
      for (int s = 0; s <= t; ++s) rs += Ps[t * LSTR + s];
      float qn = 0.f;
      for (int d = 0; d < 64; ++d) qn += Qb[t * LSTR + d] * nS[d];
      dnL[t] = 1.0f / fmaxf(gI[t] * qn + rs, EPSV);
      uL[t]  = __expf(aL[t] + bL[63] - mL[63]);
    }
    __syncthreads();

    // ---- P4: H = (gI .* (Q C^T) + P V) * (1/den) ----
    {
      v8f hInter = {0.f, 0.f, 0.f, 0.f, 0.f, 0.f, 0.f, 0.f};
      #pragma unroll
      for (int kd = 0; kd < 16; ++kd) {
        int kk = kd * 4 + 2 * kh;
        v2f a = *(const v2f*)&Qb[(tm * 16 + l16) * LSTR + kk];   // A = Q
        v2f b = *(const v2f*)&Cs[(tn * 16 + l16) * LSTR + kk];   // B = C^T
        hInter = wmma_f32(a, b, hInter);
      }
      v8f hIntra = {0.f, 0.f, 0.f, 0.f, 0.f, 0.f, 0.f, 0.f};
      for (int kd = 0; kd < 4 * (tm + 1); ++kd) {                // P lower-tri: shorter K
        int kk = kd * 4 + 2 * kh;
        v2f a = *(const v2f*)&Ps[(tm * 16 + l16) * LSTR + kk];   // A = P
        v2f b;
        b.x = Vb[kk * LSTR + tn * 16 + l16];                     // B = V (row-major)
        b.y = Vb[(kk + 1) * LSTR + tn * 16 + l16];
        hIntra = wmma_f32(a, b, hIntra);
      }
      #pragma unroll
      for (int r = 0; r < 8; ++r) {
        int t = tm * 16 + r + 8 * kh;
        int a = tn * 16 + l16;
        oB[(size_t)(c0 + t) * 64 + a] = (gI[t] * hInter[r] + hIntra[r]) * dnL[t];
      }
    }
    __syncthreads();

    // ---- P5: state update  C = gC*C + (diag(u)V)^T K ;  n = gC*n + K^T u ----
    {
      float gC = gI[63];
      v8f acc;
      #pragma unroll
      for (int r = 0; r < 8; ++r) {
        int ar = tm * 16 + r + 8 * kh;
        int bc = tn * 16 + l16;
        acc[r] = gC * Cs[ar * LSTR + bc];
      }
      #pragma unroll
      for (int kd = 0; kd < 16; ++kd) {
        int kk = kd * 4 + 2 * kh;
        v2f a, b;
        a.x = uL[kk]     * Vb[kk * LSTR       + tm * 16 + l16];  // A = (diag(u)V)^T
        a.y = uL[kk + 1] * Vb[(kk + 1) * LSTR + tm * 16 + l16];
        b.x = Kb[kk * LSTR       + tn * 16 + l16];               // B = K
        b.y = Kb[(kk + 1) * LSTR + tn * 16 + l16];
        acc = wmma_f32(a, b, acc);
      }
      #pragma unroll
      for (int r = 0; r < 8; ++r) {
        int ar = tm * 16 + r + 8 * kh;
        int bc = tn * 16 + l16;
        Cs[ar * LSTR + bc] = acc[r];
      }
      if (tid < 64) {
        float nv = gC * nS[tid];
        for (int s = 0; s < 64; ++s) nv += uL[s] * Kb[s * LSTR + tid];
        nS[tid] = nv;
      }
    }
    __syncthreads();
  }
}

extern "C" void kernel_launch(void* const* d_in, const int* in_sizes, int n_in,
                              void* d_out, int out_size, void* d_ws, size_t ws_size,
                              hipStream_t stream) {
  const float* q  = (const float*)d_in[0];
  const float* k  = (const float*)d_in[1];
  const float* v  = (const float*)d_in[2];
  const float* ig = (const float*)d_in[3];
  const float* fg = (const float*)d_in[4];
  float* out = (float*)d_out;
  const int BH = 32;                 // B*NH = 4*8
  const int S  = in_sizes[3] / BH;   // i has shape [B,NH,S]
  mlstm_fwd<<<dim3(BH), dim3(512), 0, stream>>>(q, k, v, ig, fg, out, S);
}